// LowpassFilter_34531537060274
// MI455X (gfx1250) — compile-verified
//
#include <hip/hip_runtime.h>
#include <hip/hip_bf16.h>
#include <math.h>

// Biquad lowpass over [128][160000] f32, chunked-parallel IIR decomposition.
//
// Phase 1: per (channel, chunk) zero-state particular-solution end-state.
// Phase 2: per channel, wave-parallel Kogge-Stone scan of the 2x2 affine
//          boundary-state recurrence (16 chunks/lane + 5 log-steps).
// Phase 3: per (channel-group of 16, chunk), blocked recurrence as 16x16 GEMMs
//          on V_WMMA_F32_16X16X4_F32 (4 chained WMMAs per 16-step block).

typedef __attribute__((ext_vector_type(2))) float v2f;
typedef __attribute__((ext_vector_type(4))) float v4f;
typedef __attribute__((ext_vector_type(8))) float v8f;

#define T_LEN   160000
#define NCHAN   128
#define CHUNK   320          // samples per chunk (multiple of 16; 320 = 5*2^6)
#define NCHUNK  500          // T_LEN / CHUNK
#define NBLK    20           // CHUNK / 16
#define SEG     16           // chunks per lane in phase-2 scan (16*32 >= 500)

// ---------------------------------------------------------------------------
// Phase 1: zero-initial-state end-state per (channel, chunk).
// One thread per (channel, chunk): 320 serial steps, float4 input loads.
// summ[(ch*NCHUNK + k)*2 + {0,1}] = { y[C-1], y[C-2] } with y-init = 0.
// ---------------------------------------------------------------------------
__global__ void biquad_phase1(const float* __restrict__ X,
                              float* __restrict__ summ,
                              float b0, float b1, float b2, float a1, float a2)
{
    int tid = blockIdx.x * blockDim.x + threadIdx.x;
    if (tid >= NCHAN * NCHUNK) return;
    int ch = tid / NCHUNK;
    int k  = tid - ch * NCHUNK;

    const float* xc = X + (size_t)ch * T_LEN + (size_t)k * CHUNK;

    float xm1 = 0.f, xm2 = 0.f;
    if (k > 0) { xm1 = xc[-1]; xm2 = xc[-2]; }

    float y1 = 0.f, y2 = 0.f;   // y[n-1], y[n-2]
#pragma unroll 4
    for (int i = 0; i < CHUNK; i += 4) {
        // WGP-scope prefetch ~512B ahead on this lane's private stream.
        __builtin_prefetch(xc + i + 128, 0, 3);
        v4f x4 = *(const v4f*)(xc + i);
        float f0 = b0 * x4.x + b1 * xm1  + b2 * xm2;
        float f1 = b0 * x4.y + b1 * x4.x + b2 * xm1;
        float f2 = b0 * x4.z + b1 * x4.y + b2 * x4.x;
        float f3 = b0 * x4.w + b1 * x4.z + b2 * x4.y;
        float ya = f0 - a1 * y1 - a2 * y2;
        float yb = f1 - a1 * ya - a2 * y1;
        float yc = f2 - a1 * yb - a2 * ya;
        float yd = f3 - a1 * yc - a2 * yb;
        y2 = yc; y1 = yd;
        xm1 = x4.w; xm2 = x4.z;
    }
    summ[(size_t)tid * 2 + 0] = y1;   // y[C-1]
    summ[(size_t)tid * 2 + 1] = y2;   // y[C-2]
}

// ---------------------------------------------------------------------------
// Phase 2: one wave per channel. Boundary recurrence s_k = P*s_{k-1} + d_k
// (P = M^CHUNK, M = [[-a1,-a2],[1,0]]) solved as a parallel scan of affine
// maps: 16 sequential chunks per lane, then Kogge-Stone across 32 lanes.
// bound[(ch*NCHUNK + k)*2 + {0,1}] = state entering chunk k.
// ---------------------------------------------------------------------------
__global__ void __launch_bounds__(32)
biquad_phase2(const float* __restrict__ summ,
              float* __restrict__ bound,
              float a1, float a2)
{
    int ch   = blockIdx.x;        // 0..127
    int lane = threadIdx.x;       // 0..31

    // --- P = M^CHUNK via M^5 then 6 squarings (320 = 5 * 2^6) -------------
    float p1 = 1.f, p2 = 0.f;     // image of e1
    float q1 = 0.f, q2 = 1.f;     // image of e2
#pragma unroll
    for (int i = 0; i < 5; ++i) {
        float np = -a1 * p1 - a2 * p2; p2 = p1; p1 = np;
        float nq = -a1 * q1 - a2 * q2; q2 = q1; q1 = nq;
    }
#pragma unroll
    for (int s = 0; s < 6; ++s) { // square: [[p1 q1],[p2 q2]]^2
        float n11 = p1 * p1 + q1 * p2;
        float n12 = p1 * q1 + q1 * q2;
        float n21 = p2 * p1 + q2 * p2;
        float n22 = p2 * q1 + q2 * q2;
        p1 = n11; q1 = n12; p2 = n21; q2 = n22;
    }

    const float* sc = summ  + (size_t)ch * NCHUNK * 2;
    float*       bc = bound + (size_t)ch * NCHUNK * 2;
    int q0 = lane * SEG;

    // --- Pass 1: per-lane segment aggregate (matrix part = P^SEG) ---------
    float w1 = 0.f, w2 = 0.f;
#pragma unroll 4
    for (int i = 0; i < SEG; ++i) {
        int q = q0 + i;
        float d1 = 0.f, d2 = 0.f;
        if (q < NCHUNK) { d1 = sc[2 * q]; d2 = sc[2 * q + 1]; }
        float nw1 = p1 * w1 + q1 * w2 + d1;
        float nw2 = p2 * w1 + q2 * w2 + d2;
        w1 = nw1; w2 = nw2;
    }
    // Per-lane matrix starts at P^SEG (4 squarings: 16 = 2^4).
    float m11 = p1, m12 = q1, m21 = p2, m22 = q2;
#pragma unroll
    for (int s = 0; s < 4; ++s) {
        float n11 = m11 * m11 + m12 * m21;
        float n12 = m11 * m12 + m12 * m22;
        float n21 = m21 * m11 + m22 * m21;
        float n22 = m21 * m12 + m22 * m22;
        m11 = n11; m12 = n12; m21 = n21; m22 = n22;
    }

    // --- Kogge-Stone inclusive scan of (M, w), composing left-to-right ----
#pragma unroll
    for (int o = 1; o < 32; o <<= 1) {
        float o11 = __shfl_up(m11, o, 32);
        float o12 = __shfl_up(m12, o, 32);
        float o21 = __shfl_up(m21, o, 32);
        float o22 = __shfl_up(m22, o, 32);
        float ow1 = __shfl_up(w1, o, 32);
        float ow2 = __shfl_up(w2, o, 32);
        if (lane >= o) {
            float nw1 = m11 * ow1 + m12 * ow2 + w1;   // apply other first
            float nw2 = m21 * ow1 + m22 * ow2 + w2;
            w1 = nw1; w2 = nw2;
            float n11 = m11 * o11 + m12 * o21;
            float n12 = m11 * o12 + m12 * o22;
            float n21 = m21 * o11 + m22 * o21;
            float n22 = m21 * o12 + m22 * o22;
            m11 = n11; m12 = n12; m21 = n21; m22 = n22;
        }
    }
    // Incoming state for this lane's segment = inclusive result of lane-1.
    float in1 = __shfl_up(w1, 1, 32);
    float in2 = __shfl_up(w2, 1, 32);
    if (lane == 0) { in1 = 0.f; in2 = 0.f; }

    // --- Pass 2: emit per-chunk incoming states ---------------------------
    float s1 = in1, s2 = in2;
#pragma unroll 4
    for (int i = 0; i < SEG; ++i) {
        int q = q0 + i;
        if (q < NCHUNK) {
            bc[2 * q + 0] = s1;
            bc[2 * q + 1] = s2;
            float d1 = sc[2 * q], d2 = sc[2 * q + 1];
            float ns1 = p1 * s1 + q1 * s2 + d1;
            float ns2 = p2 * s1 + q2 * s2 + d2;
            s1 = ns1; s2 = ns2;
        }
    }
}

// ---------------------------------------------------------------------------
// Phase 3: one wave = 16 channels x one chunk. Each 16-step block:
//   Y(16x16) = L(16x16) * F(16x16) + u*y_in1 + v*y_in2
// via 4x V_WMMA_F32_16X16X4_F32 (K=16 in 4 sub-K slabs).
// A layout (16x4 f32, 2 VGPRs): lanes 0-15 hold {K=0,K=1}, lanes 16-31 {K=2,K=3}.
// B layout (4x16 f32, 2 VGPRs): lanes 0-15 rows {0,1}, lanes 16-31 rows {2,3}.
// C/D layout: VGPR r -> time row r (lanes 0-15) / r+8 (lanes 16-31), N = lane&15.
// ---------------------------------------------------------------------------
__global__ void __launch_bounds__(256)
biquad_phase3(const float* __restrict__ X,
              const float* __restrict__ bound,
              float* __restrict__ Y,
              float b0, float b1, float b2, float a1, float a2)
{
    __shared__ float sh[17];           // h[0..16], AR impulse response

    int lane = threadIdx.x & 31;
    int wave = threadIdx.x >> 5;       // channel group 0..7
    int k    = blockIdx.x;             // chunk 0..NCHUNK-1

    if (threadIdx.x == 0) {
        float h0 = 1.0f, h1 = -a1;
        sh[0] = h0; sh[1] = h1;
        for (int i = 2; i <= 16; ++i) {
            float hn = -a1 * h1 - a2 * h0;
            sh[i] = hn; h0 = h1; h1 = hn;
        }
    }
    __syncthreads();

    int m    = lane & 15;              // matrix row (A) / column N (B,C,D)
    int half = lane >> 4;

    // A sub-matrices: A[j] covers K = 4j..4j+3 of L[m][k] = h[m-k] (lower-tri).
    v2f A[4];
#pragma unroll
    for (int j = 0; j < 4; ++j) {
        int k0 = 4 * j + 2 * half;
        A[j].x = (m >= k0)     ? sh[m - k0]     : 0.0f;
        A[j].y = (m >= k0 + 1) ? sh[m - k0 - 1] : 0.0f;
    }
    // State-injection rows: u[n] = h[n+1], v[n] = -a2*h[n] ; per-lane-half row.
    float uu[8], vv[8];
#pragma unroll
    for (int r = 0; r < 8; ++r) {
        int mm = r + 8 * half;
        uu[r] = sh[mm + 1];
        vv[r] = -a2 * sh[mm];
    }

    int ch = wave * 16 + m;
    const float* xc = X + (size_t)ch * T_LEN;
    float*       yc = Y + (size_t)ch * T_LEN;
    int tbase = k * CHUNK;

    float y1 = bound[((size_t)ch * NCHUNK + k) * 2 + 0];  // y[-1]
    float y2 = bound[((size_t)ch * NCHUNK + k) * 2 + 1];  // y[-2]
    float xm1 = 0.f, xm2 = 0.f;
    if (k > 0) { xm1 = xc[tbase - 1]; xm2 = xc[tbase - 2]; }

#pragma unroll 1
    for (int b = 0; b < NBLK; ++b) {
        int t = tbase + b * 16;

        // Accumulator seeded with incoming-state contribution.
        v8f c;
#pragma unroll
        for (int r = 0; r < 8; ++r) c[r] = uu[r] * y1 + vv[r] * y2;

#pragma unroll
        for (int j = 0; j < 4; ++j) {
            float f0 = 0.f, f1 = 0.f, f2 = 0.f, f3 = 0.f;
            if (half == 0) {                       // lanes 0-15 load + FIR
                if (j == 0)   // WGP-scope prefetch 4 blocks ahead
                    __builtin_prefetch(xc + t + 64, 0, 3);
                v4f x4 = *(const v4f*)(xc + t + 4 * j);
                f0 = b0 * x4.x + b1 * xm1  + b2 * xm2;
                f1 = b0 * x4.y + b1 * x4.x + b2 * xm1;
                f2 = b0 * x4.z + b1 * x4.y + b2 * x4.x;
                f3 = b0 * x4.w + b1 * x4.z + b2 * x4.y;
                xm1 = x4.w; xm2 = x4.z;
            }
            // Rows 2,3 of F live in lanes 16-31: fetch from the owning lane.
            float f2s = __shfl(f2, m, 32);
            float f3s = __shfl(f3, m, 32);
            v2f Bv;
            Bv.x = half ? f2s : f0;
            Bv.y = half ? f3s : f1;
            // D = A*B + C  (EXEC all-ones here)
            c = __builtin_amdgcn_wmma_f32_16x16x4_f32(
                    false, A[j], false, Bv, (short)0, c, false, false);
        }

        // Store Y block: VGPR r -> time t + r + 8*half for channel ch.
#pragma unroll
        for (int r = 0; r < 8; ++r)
            yc[t + r + 8 * half] = c[r];

        // Next block's state: y[15] = c[7] (upper half), y[14] = c[6] (upper).
        y1 = __shfl(c[7], m + 16, 32);
        y2 = __shfl(c[6], m + 16, 32);
    }
}

// ---------------------------------------------------------------------------
extern "C" void kernel_launch(void* const* d_in, const int* in_sizes, int n_in,
                              void* d_out, int out_size, void* d_ws, size_t ws_size,
                              hipStream_t stream) {
    const float* clip = (const float*)d_in[0];
    float* out = (float*)d_out;

    // Coefficients: double math then f32 cast, mirroring the reference.
    const double SR = 32000.0, CUT = 8000.0, Qf = 0.707;
    double w0    = 2.0 * M_PI * CUT / SR;
    double alpha = sin(w0) / (2.0 * Qf);
    double cw    = cos(w0);
    double b0d = (1.0 - cw) / 2.0, b1d = 1.0 - cw, b2d = b0d;
    double a0d = 1.0 + alpha, a1d = -2.0 * cw, a2d = 1.0 - alpha;
    float b0 = (float)(b0d / a0d), b1 = (float)(b1d / a0d), b2 = (float)(b2d / a0d);
    float a1 = (float)(a1d / a0d), a2 = (float)(a2d / a0d);

    // Workspace: [NCHAN*NCHUNK*2] summaries + [NCHAN*NCHUNK*2] boundary states.
    float* summ  = (float*)d_ws;                      // 128000 floats (512 KB)
    float* bound = summ + (size_t)NCHAN * NCHUNK * 2; // 128000 floats (512 KB)
    (void)ws_size; // requires >= 1 MB

    {   // Phase 1: 64000 threads (2000 waves)
        int total = NCHAN * NCHUNK;
        dim3 grid((total + 255) / 256), block(256);
        biquad_phase1<<<grid, block, 0, stream>>>(clip, summ, b0, b1, b2, a1, a2);
    }
    {   // Phase 2: one wave per channel, scan-parallel boundary pass
        biquad_phase2<<<dim3(NCHAN), dim3(32), 0, stream>>>(summ, bound, a1, a2);
    }
    {   // Phase 3: NCHUNK blocks x 8 waves (4000 waves of WMMA work)
        biquad_phase3<<<dim3(NCHUNK), dim3(256), 0, stream>>>(
            clip, bound, out, b0, b1, b2, a1, a2);
    }
}